// build_fpn_mask_graph_29480655520198
// MI455X (gfx1250) — compile-verified
//
#include <hip/hip_runtime.h>
#include <math.h>

#define POOL 7
#define NBIN (POOL * POOL)   // 49
#define CC   16              // channels per block (power of 2)
#define BLK  256
#define MAXELEM   1022       // max patch elements per channel that we stage
#define MAXSTRIDE 1024       // LDS channel stride cap (words); CC*MAXSTRIDE*4 = 64 KB

// Explicit LDS address space so reads lower to ds_load_* (not flat_load)
typedef __attribute__((address_space(3))) float lds_float;

// ---- CDNA5 async global -> LDS copy (per-lane), tracked by ASYNCcnt ----
__device__ __forceinline__ void async_g2l_b32(unsigned lds_byte_off, const float* gaddr) {
    asm volatile("global_load_async_to_lds_b32 %0, %1, off"
                 :: "v"(lds_byte_off), "v"(gaddr)
                 : "memory");
}
__device__ __forceinline__ void wait_async0() {
    asm volatile("s_wait_asynccnt 0" ::: "memory");
}

// Exact unsigned division by runtime d via round-up magic: valid for n,d < 2^16.
// M = ceil(2^32/d); floor(n*M / 2^32) == n/d because n*(d - 2^32 mod d) < 2^32.
__device__ __forceinline__ unsigned magic_of(unsigned d)              { return 0xFFFFFFFFu / d + 1u; }
__device__ __forceinline__ unsigned magic_div(unsigned n, unsigned M) { return __umulhi(n, M); }

__global__ __launch_bounds__(BLK)
void roialign_fpn_cdna5_kernel(const float* __restrict__ p2, const float* __restrict__ p3,
                               const float* __restrict__ p4, const float* __restrict__ p5,
                               const float* __restrict__ boxes, const int* __restrict__ bidx,
                               float* __restrict__ out, int N)
{
    __shared__ float patch[CC * MAXSTRIDE];   // 64 KB

    const int n   = blockIdx.x;
    const int c0  = blockIdx.y * CC;
    const int tid = threadIdx.x;
    if (n >= N) return;

    // ---------- uniform per-box setup (scalarizes) ----------
    const float bx1 = boxes[n * 4 + 0], by1 = boxes[n * 4 + 1];
    const float bx2 = boxes[n * 4 + 2], by2 = boxes[n * 4 + 3];
    const int   b   = bidx[n];

    const float area = (by2 - by1) * (bx2 - bx1);
    int level = (int)rintf(0.5f * log2f(area)) + 4;   // rint == round-half-even (matches jnp.round)
    level = level < 0 ? 0 : (level > 3 ? 3 : level);

    const float* feat; int W;
    if      (level == 0) { feat = p2; W = 256; }
    else if (level == 1) { feat = p3; W = 128; }
    else if (level == 2) { feat = p4; W = 64;  }
    else                 { feat = p5; W = 32;  }
    const int H = W;
    const int C = 256;
    const size_t planeStride = (size_t)H * W;

    const float sw  = (float)W;           // reference scales box by f.shape[3] for BOTH dims
    const float x1s = bx1 * sw, y1s = by1 * sw;
    const float x2s = bx2 * sw, y2s = by2 * sw;
    const float roi_w = fmaxf(x2s - x1s, 1.0f);
    const float roi_h = fmaxf(y2s - y1s, 1.0f);
    const float bin_w = roi_w * (1.0f / POOL);
    const float bin_h = roi_h * (1.0f / POOL);

    // ---------- patch bounds: rows/cols touched by the (monotone) sample grid ----------
    // first sample offset = 0.25*bin, last = 6.75*bin (POOL=7, SR=2)
    float yf = fminf(fmaxf(y1s + 0.25f * bin_h, 0.0f), (float)(H - 1));
    float yl = fminf(fmaxf(y1s + 6.75f * bin_h, 0.0f), (float)(H - 1));
    const int y0 = (int)floorf(yf);
    const int PH = min((int)floorf(yl) + 1, H - 1) - y0 + 1;

    float xf = fminf(fmaxf(x1s + 0.25f * bin_w, 0.0f), (float)(W - 1));
    float xl = fminf(fmaxf(x1s + 6.75f * bin_w, 0.0f), (float)(W - 1));
    const int x0 = (int)floorf(xf);
    const int PW = min((int)floorf(xl) + 1, W - 1) - x0 + 1;

    const int elems      = PH * PW;
    const int chanStride = elems | 1;                  // odd stride helps spread 64 LDS banks
    const bool staged    = (elems <= MAXELEM);         // block-uniform condition

    // ---------- stage patch into LDS with async global->LDS loads ----------
    if (staged) {
        const float*   gbase    = feat + ((size_t)(b * C + c0) * H + y0) * W + x0;
        const unsigned smemBase = (unsigned)(size_t)(void*)patch;  // low 32 bits = LDS byte offset
        const unsigned Me       = magic_of((unsigned)elems);       // scalar, once per block
        const unsigned Mw       = magic_of((unsigned)PW);
        const unsigned total    = (unsigned)(CC * elems);          // <= 16352 < 2^16
        for (unsigned i = tid; i < total; i += BLK) {
            const unsigned cc  = magic_div(i, Me);                 // i / elems   (exact)
            const unsigned rem = i - cc * (unsigned)elems;         // = r*PW + x  (matches LDS layout)
            const unsigned r   = magic_div(rem, Mw);               // rem / PW    (exact)
            const unsigned x   = rem - r * (unsigned)PW;
            const float* g = gbase + (size_t)cc * planeStride + (size_t)r * W + x;
            async_g2l_b32(smemBase + 4u * (cc * (unsigned)chanStride + rem), g);
        }
        wait_async0();        // this wave's async copies landed in LDS
        __syncthreads();      // make all waves' copies visible
    }

    // ---------- bilinear sampling ----------
    // Mapping i = c*49 + bin: a wave's 32 lanes cover 32 consecutive i (bins, possibly of two
    // channels) -> the 32 output floats per wave-store are CONTIGUOUS (coalesced).
    for (int i = tid; i < CC * NBIN; i += BLK) {
        const int c   = i / NBIN;                      // compile-time magic division
        const int bin = i - c * NBIN;
        const int byi = bin / POOL;
        const int bxi = bin - byi * POOL;

        const float*     gplane = feat + (size_t)(b * C + c0 + c) * planeStride;
        const lds_float* lplane = (const lds_float*)(patch) + c * chanStride;

        float acc = 0.0f;
        #pragma unroll
        for (int sy = 0; sy < 2; ++sy) {
            const float yy  = y1s + bin_h * ((float)byi + (sy ? 0.75f : 0.25f));
            const bool  vy  = (yy > -1.0f) && (yy < (float)H);
            const float ycl = fminf(fmaxf(yy, 0.0f), (float)(H - 1));
            const int   ylo = (int)floorf(ycl);
            const int   yhi = min(ylo + 1, H - 1);
            const float fy  = ycl - (float)ylo;
            #pragma unroll
            for (int sx = 0; sx < 2; ++sx) {
                const float xx  = x1s + bin_w * ((float)bxi + (sx ? 0.75f : 0.25f));
                const bool  vx  = (xx > -1.0f) && (xx < (float)W);
                const float xcl = fminf(fmaxf(xx, 0.0f), (float)(W - 1));
                const int   xlo = (int)floorf(xcl);
                const int   xhi = min(xlo + 1, W - 1);
                const float fx  = xcl - (float)xlo;

                float v00, v01, v10, v11;
                if (staged) {
                    const int r0 = ylo - y0, r1 = yhi - y0;
                    const int q0 = xlo - x0, q1 = xhi - x0;
                    v00 = lplane[r0 * PW + q0];        // ds_load_b32
                    v01 = lplane[r0 * PW + q1];
                    v10 = lplane[r1 * PW + q0];
                    v11 = lplane[r1 * PW + q1];
                } else {
                    v00 = gplane[(size_t)ylo * W + xlo];
                    v01 = gplane[(size_t)ylo * W + xhi];
                    v10 = gplane[(size_t)yhi * W + xlo];
                    v11 = gplane[(size_t)yhi * W + xhi];
                }
                const float val = (1.0f - fy) * ((1.0f - fx) * v00 + fx * v01)
                                +         fy  * ((1.0f - fx) * v10 + fx * v11);
                acc += (vy && vx) ? val : 0.0f;
            }
        }
        // out[n][c0+c][by][bx]; bin index is contiguous across lanes
        out[((size_t)n * C + (c0 + c)) * NBIN + bin] = acc * 0.25f;
    }
}

extern "C" void kernel_launch(void* const* d_in, const int* in_sizes, int n_in,
                              void* d_out, int out_size, void* d_ws, size_t ws_size,
                              hipStream_t stream) {
    const float* p2    = (const float*)d_in[0];
    const float* p3    = (const float*)d_in[1];
    const float* p4    = (const float*)d_in[2];
    const float* p5    = (const float*)d_in[3];
    const float* boxes = (const float*)d_in[4];
    const int*   bidx  = (const int*)d_in[5];
    float*       out   = (float*)d_out;
    const int N = in_sizes[5];                 // box_index count

    dim3 grid((unsigned)N, 256 / CC);
    roialign_fpn_cdna5_kernel<<<grid, BLK, 0, stream>>>(p2, p3, p4, p5, boxes, bidx, out, N);
}